// GNNModel_52793738002724
// MI455X (gfx1250) — compile-verified
//
#include <hip/hip_runtime.h>
#include <hip/hip_bf16.h>

typedef float v2f __attribute__((ext_vector_type(2)));
typedef float v8f __attribute__((ext_vector_type(8)));

// Problem constants (from reference)
constexpr int B_    = 4;
constexpr int N_    = 10000;
constexpr int E_    = 320000;
constexpr int D_IN  = 128;
constexpr int D_H   = 128;
constexpr int D_OUT = 64;

// ---------------- degree / norm prep ----------------

__global__ __launch_bounds__(256) void init_deg(float* deg) {
    int n = blockIdx.x * blockDim.x + threadIdx.x;
    if (n < N_) deg[n] = 1.0f;               // self-loop weight
}

__global__ __launch_bounds__(256) void deg_accum(const long long* __restrict__ dst,
                                                 const float* __restrict__ w,
                                                 float* deg) {
    int e = blockIdx.x * blockDim.x + threadIdx.x;
    if (e < E_) atomicAdd(&deg[(int)dst[e]], w[e]);
}

__global__ __launch_bounds__(256) void dinv_kernel(const float* __restrict__ deg,
                                                   float* dinv, float* s) {
    int n = blockIdx.x * blockDim.x + threadIdx.x;
    if (n >= N_) return;
    float d  = deg[n];
    float di = (d > 0.0f) ? rsqrtf(d) : 0.0f;
    dinv[n]  = di;
    s[n]     = di * di;                       // self-loop norm contribution to src-sum
}

__global__ __launch_bounds__(256) void s_accum(const long long* __restrict__ src,
                                               const long long* __restrict__ dst,
                                               const float* __restrict__ w,
                                               const float* __restrict__ dinv,
                                               float* s) {
    int e = blockIdx.x * blockDim.x + threadIdx.x;
    if (e >= E_) return;
    int si = (int)src[e], di = (int)dst[e];
    atomicAdd(&s[si], dinv[si] * w[e] * dinv[di]);
}

// ---------------- W1 fragment pre-pack ----------------
// For column-tile t (0..7), k-step ks (0..31), lane l (0..31):
//   half = l>>4, idx = l&15, kk = 4*ks + 2*half, col = 16*t + idx
//   Wp[(t*32+ks)*32 + l] = { W[kk][col], W[kk+1][col] }
// Inner GEMM loop then issues one coalesced b64 load per WMMA.

__global__ __launch_bounds__(256) void pack_W(const float* __restrict__ W,
                                              float2* __restrict__ Wp) {
    int i = blockIdx.x * blockDim.x + threadIdx.x;      // 8*32*32 = 8192
    if (i >= (D_H / 16) * 32 * 32) return;
    int l  = i & 31;
    int ks = (i >> 5) & 31;
    int t  = i >> 10;
    int half = l >> 4, idx = l & 15;
    int kk  = 4 * ks + 2 * half;
    int col = 16 * t + idx;
    Wp[i] = make_float2(W[(size_t)kk * D_H + col], W[(size_t)(kk + 1) * D_H + col]);
}

// ---------------- WMMA fp32 GEMM: XW = X(M x 128) * W(128 x 128) ----------------
// One block = 16-row strip; 8 waves, each computes one 16x16 C tile.
// A tile staged in LDS (coalesced float4 -> ds_load_2addr_b64 frags),
// B frags streamed as packed b64 loads (L1-resident, identical across blocks).

__global__ __launch_bounds__(256) void gemm1_wmma(const float* __restrict__ X,
                                                  const float2* __restrict__ Wp,
                                                  float* __restrict__ XW) {
    constexpr int K = D_IN, NC = D_H;
    __shared__ float As[16 * K];              // 8 KB
    const int row0 = blockIdx.x * 16;
    const int wave = threadIdx.x >> 5;        // 0..7 -> column tile
    const int lane = threadIdx.x & 31;

    // cooperative coalesced load of the 16 x 128 A tile
    {
        const float4* srcp = (const float4*)(X + (size_t)row0 * K);
        float4*       dstp = (float4*)As;
        for (int i = threadIdx.x; i < 16 * K / 4; i += blockDim.x) dstp[i] = srcp[i];
    }
    __syncthreads();

    const int half = lane >> 4;               // K-pair selector
    const int idx  = lane & 15;               // row (A) / col (B) within tile
    const int col0 = wave * 16;

    const float2* wp = Wp + (size_t)wave * 32 * 32 + lane;

    v8f c = {};
#pragma unroll
    for (int ks = 0; ks < K / 4; ++ks) {
        const int kk = 4 * ks + 2 * half;
        v2f a, b;
        a.x = As[idx * K + kk];
        a.y = As[idx * K + kk + 1];
        const float2 bw = wp[(size_t)ks * 32];
        b.x = bw.x;
        b.y = bw.y;
        c = __builtin_amdgcn_wmma_f32_16x16x4_f32(
                /*neg_a=*/false, a, /*neg_b=*/false, b,
                /*c_mod=*/(short)0, c, /*reuse_a=*/false, /*reuse_b=*/false);
    }

    // C/D layout: VGPR v -> row v + 8*half, col idx
    float* out = XW + (size_t)row0 * NC + col0;
#pragma unroll
    for (int v = 0; v < 8; ++v)
        out[(size_t)(v + 8 * half) * NC + idx] = c[v];
}

// ---------------- edge scatter, layer 1 (one wave per edge) ----------------

__global__ __launch_bounds__(256) void scatter1(const long long* __restrict__ src,
                                                const long long* __restrict__ dst,
                                                const float* __restrict__ w,
                                                const float* __restrict__ dinv,
                                                const float* __restrict__ xw,
                                                float* agg) {
    const int e    = (blockIdx.x * blockDim.x + threadIdx.x) >> 5;
    const int lane = threadIdx.x & 31;
    if (e >= E_) return;
    const int si = (int)src[e], di = (int)dst[e];
    const float nrm = dinv[si] * w[e] * dinv[di];
#pragma unroll
    for (int b = 0; b < B_; ++b) {
        const float4 v = *(const float4*)(xw + ((size_t)b * N_ + si) * D_H + lane * 4);
        float* ap = agg + ((size_t)b * N_ + di) * D_H + lane * 4;
        atomicAdd(ap + 0, v.x * nrm);
        atomicAdd(ap + 1, v.y * nrm);
        atomicAdd(ap + 2, v.z * nrm);
        atomicAdd(ap + 3, v.w * nrm);
    }
}

// ---------------- fused self-loop + bias + ReLU (in place on agg) ----------------

__global__ __launch_bounds__(256) void bias_relu_self(float* agg,
                                                      const float* __restrict__ xw,
                                                      const float* __restrict__ dinv,
                                                      const float* __restrict__ b1) {
    const size_t total4 = (size_t)B_ * N_ * D_H / 4;
    size_t i = (size_t)blockIdx.x * blockDim.x + threadIdx.x;
    if (i >= total4) return;
    const int d4 = (int)(i % (D_H / 4));
    const size_t bn = i / (D_H / 4);
    const int n  = (int)(bn % N_);
    const float di = dinv[n];
    const float sl = di * di;                 // self-loop norm
    float4 a  = ((float4*)agg)[i];
    const float4 x  = ((const float4*)xw)[i];
    const float4 bb = ((const float4*)b1)[d4];
    a.x = fmaxf(fmaf(sl, x.x, a.x) + bb.x, 0.0f);
    a.y = fmaxf(fmaf(sl, x.y, a.y) + bb.y, 0.0f);
    a.z = fmaxf(fmaf(sl, x.z, a.z) + bb.z, 0.0f);
    a.w = fmaxf(fmaf(sl, x.w, a.w) + bb.w, 0.0f);
    ((float4*)agg)[i] = a;
}

// ---------------- weighted reduction r[b,:] = sum_n s[n] * h[b,n,:] ----------------

constexpr int CHUNK = 250;                    // 40 chunks cover N=10000

__global__ __launch_bounds__(128) void reduce_r(const float* __restrict__ h,
                                                const float* __restrict__ s,
                                                float* r) {
    const int b  = blockIdx.y;
    const int d  = threadIdx.x;               // 0..127
    const int n0 = blockIdx.x * CHUNK;
    const int nn = min(CHUNK, N_ - n0);
    const float* hp = h + ((size_t)b * N_ + n0) * D_H + d;
    float acc = 0.0f;
    for (int i = 0; i < nn; ++i)
        acc = fmaf(s[n0 + i], hp[(size_t)i * D_H], acc);
    atomicAdd(&r[b * D_H + d], acc);
}

// ---------------- out[b,:] = r[b,:] @ W2 / N + b2 ----------------

__global__ __launch_bounds__(256) void final_kernel(const float* __restrict__ r,
                                                    const float* __restrict__ W2,
                                                    const float* __restrict__ b2,
                                                    float* out) {
    const int t = threadIdx.x;                // 256 = 4 * 64
    const int b = t >> 6, dout = t & 63;
    float acc = 0.0f;
#pragma unroll 4
    for (int d = 0; d < D_H; ++d)
        acc = fmaf(r[b * D_H + d], W2[d * D_OUT + dout], acc);
    out[b * D_OUT + dout] = acc * (1.0f / (float)N_) + b2[dout];
}

// ---------------- launch ----------------

extern "C" void kernel_launch(void* const* d_in, const int* in_sizes, int n_in,
                              void* d_out, int out_size, void* d_ws, size_t ws_size,
                              hipStream_t stream) {
    const float*     x   = (const float*)d_in[0];            // (B,N,128)
    const long long* ei  = (const long long*)d_in[1];        // (2,E) int64
    const float*     ef  = (const float*)d_in[2];            // (E,1)
    const float*     W1  = (const float*)d_in[3];            // (128,128)
    const float*     b1  = (const float*)d_in[4];            // (128)
    const float*     W2  = (const float*)d_in[5];            // (128,64)
    const float*     b2  = (const float*)d_in[6];            // (64)
    float*           out = (float*)d_out;                    // (B,64)

    const long long* src = ei;
    const long long* dst = ei + E_;

    // workspace layout (floats)
    float*  f    = (float*)d_ws;
    float*  deg  = f;                                        // N
    float*  dinv = f + 16384;                                // N
    float*  s    = f + 32768;                                // N
    float*  xw   = f + 49152;                                // B*N*128
    float*  agg  = xw + (size_t)B_ * N_ * D_H;               // B*N*128 (becomes h)
    float*  r    = agg + (size_t)B_ * N_ * D_H;              // B*128
    float2* Wp   = (float2*)(r + B_ * D_H);                  // 8192 float2 (64 KB)

    (void)in_sizes; (void)n_in; (void)out_size; (void)ws_size;

    hipMemsetAsync(agg, 0, (size_t)B_ * N_ * D_H * sizeof(float), stream);
    hipMemsetAsync(r,   0, (size_t)B_ * D_H * sizeof(float), stream);

    init_deg <<<(N_ + 255) / 256, 256, 0, stream>>>(deg);
    deg_accum<<<(E_ + 255) / 256, 256, 0, stream>>>(dst, ef, deg);
    dinv_kernel<<<(N_ + 255) / 256, 256, 0, stream>>>(deg, dinv, s);
    s_accum  <<<(E_ + 255) / 256, 256, 0, stream>>>(src, dst, ef, dinv, s);

    // pack W1 into WMMA fragment order (once; 8192 float2)
    pack_W<<<(8192 + 255) / 256, 256, 0, stream>>>(W1, Wp);

    // xw = x @ W1 via fp32 WMMA (M = B*N = 40000 rows, 2500 row-strips)
    gemm1_wmma<<<(B_ * N_) / 16, 256, 0, stream>>>(x, Wp, xw);

    // layer-1 scatter: one wave per edge
    scatter1<<<(E_ * 32 + 255) / 256, 256, 0, stream>>>(src, dst, ef, dinv, xw, agg);

    // h = relu(agg + dinv^2 * xw + b1), in place
    {
        const size_t total4 = (size_t)B_ * N_ * D_H / 4;
        bias_relu_self<<<(unsigned)((total4 + 255) / 256), 256, 0, stream>>>(agg, xw, dinv, b1);
    }

    // r[b,:] = sum_n s[n] * h[b,n,:]
    {
        dim3 grid((N_ + CHUNK - 1) / CHUNK, B_);
        reduce_r<<<grid, 128, 0, stream>>>(agg, s, r);
    }

    // out = r @ W2 / N + b2
    final_kernel<<<1, 256, 0, stream>>>(r, W2, b2, out);
}